// newPad2d_8169027797834
// MI455X (gfx1250) — compile-verified
//
#include <hip/hip_runtime.h>

// Replicate ("edge") pad, width 2 on H and W.
// In:  (32, 256, 56, 56) fp32   Out: (32, 256, 60, 60) fp32
// Memory-bound: ~221 MB total traffic -> ~9.5 us floor at 23.3 TB/s HBM.
//
// v2 changes vs v1:
//  - Power-of-2 padded index space: idx = [plane:13][oh:6][l4:4].
//    All decomposition is shift/mask (kills two magic-division sequences,
//    ~20 VALU/thread) at the cost of ~14% idle lanes that exit immediately.
//  - Non-temporal b128 stores (output is write-once) so the 118 MB write
//    stream doesn't evict the ~103 MB input from the 192 MB L2; border
//    rows/cols are re-read and should stay L2-resident.
//  - Keep global_prefetch_b8 (gfx1250 streaming prefetch) on the read side.

#define PAD   2
#define IH    56
#define IW    56
#define OH    60
#define OW    60
#define NPLANES (32 * 256)          // 8192
#define ROW4    (OW / 4)            // 15 float4 per output row

// Padded space: 16 slots per row, 64 rows per plane.
#define IDX_TOTAL (NPLANES * 64 * 16)   // 8,388,608 -> 32768 blocks of 256

typedef float v4f __attribute__((ext_vector_type(4)));

__global__ __launch_bounds__(256) void pad_edge_f32_kernel(
    const float* __restrict__ in, float* __restrict__ out) {
  unsigned idx = blockIdx.x * blockDim.x + threadIdx.x;

  // Shift/mask decomposition (no integer division anywhere).
  int l4 = idx & 15u;           // float4 slot in row: 0..15 (15 = dead lane)
  int oh = (idx >> 4) & 63u;    // output row: 0..63 (60..63 = dead)
  int plane = idx >> 10;        // 0..8191

  if (l4 >= ROW4 || oh >= OH) return;

  // Edge replication = clamp source row.
  int ih = oh - PAD;
  ih = ih < 0 ? 0 : (ih > IH - 1 ? IH - 1 : ih);

  const float* __restrict__ irow =
      in + (size_t)plane * (IH * IW) + (size_t)ih * IW;

  int ow0 = l4 << 2;       // first output column of this float4
  int c0  = ow0 - PAD;     // unclamped source column of element 0: -2..54

  // Streaming prefetch two source rows ahead (global_prefetch_b8 on gfx1250).
  __builtin_prefetch(irow + 2 * IW + (c0 < 0 ? 0 : c0), 0, 1);

  v4f v;
  if (c0 >= 0 && c0 + 3 <= IW - 1) {
    // Interior: contiguous 4-wide read, no clamping.
    v.x = irow[c0 + 0];
    v.y = irow[c0 + 1];
    v.z = irow[c0 + 2];
    v.w = irow[c0 + 3];
  } else {
    // Border tile: branch-free per-column clamp.
    float tmp[4];
#pragma unroll
    for (int j = 0; j < 4; ++j) {
      int iw = c0 + j;
      iw = iw < 0 ? 0 : (iw > IW - 1 ? IW - 1 : iw);
      tmp[j] = irow[iw];
    }
    v.x = tmp[0]; v.y = tmp[1]; v.z = tmp[2]; v.w = tmp[3];
  }

  // Output row stride = 240 B (16B-aligned) -> aligned b128 store.
  // Non-temporal: write-once stream, keep L2 for the input.
  v4f* __restrict__ orow4 =
      (v4f*)(out + (size_t)plane * (OH * OW) + (size_t)oh * OW);
  __builtin_nontemporal_store(v, orow4 + l4);
}

extern "C" void kernel_launch(void* const* d_in, const int* in_sizes, int n_in,
                              void* d_out, int out_size, void* d_ws, size_t ws_size,
                              hipStream_t stream) {
  (void)in_sizes; (void)n_in; (void)d_ws; (void)ws_size; (void)out_size;
  const float* x = (const float*)d_in[0];
  float* y = (float*)d_out;

  const int threads = 256;                         // 8 wave32s / block
  const int blocks  = IDX_TOTAL / threads;         // 32768
  pad_edge_f32_kernel<<<blocks, threads, 0, stream>>>(x, y);
}